// PaPi_59760174957096
// MI455X (gfx1250) — compile-verified
//
#include <hip/hip_runtime.h>

// Problem sizes (fixed by the reference)
#define B_N 1024
#define C_N 10000
#define D_N 128

typedef __attribute__((ext_vector_type(2))) float v2f;
typedef __attribute__((ext_vector_type(8))) float v8f;

// ---------------------------------------------------------------------------
// Kernel 1: fused pass-through copy of output_q/output_k + weighted argmax of
// both, one float4-vectorized sweep over the big [B, C] tensors (the dominant
// ~200 MB of traffic -> global_load_b128/global_store_b128). One block/row.
// Tie-break = smallest index (matches jnp.argmax): ascending scan, strict >.
// ---------------------------------------------------------------------------
__global__ void __launch_bounds__(256) argmax_copy_kernel(
    const float* __restrict__ oq, const float* __restrict__ ok,
    const float* __restrict__ pY,
    float* __restrict__ oq_copy, float* __restrict__ ok_copy,
    int* __restrict__ pseudo_q, int* __restrict__ pseudo_k)
{
    __shared__ float svq[256];
    __shared__ int   siq[256];
    __shared__ float svk[256];
    __shared__ int   sik[256];

    const int b   = blockIdx.x;
    const int tid = threadIdx.x;
    const size_t row4 = (size_t)b * (C_N / 4);

    const float4* oq4 = (const float4*)oq + row4;
    const float4* ok4 = (const float4*)ok + row4;
    const float4* pY4 = (const float4*)pY + row4;
    float4* oqc4 = (float4*)oq_copy + row4;
    float4* okc4 = (float4*)ok_copy + row4;

    float bvq = -3.402823466e38f; int biq = C_N;
    float bvk = -3.402823466e38f; int bik = C_N;

    for (int c4 = tid; c4 < C_N / 4; c4 += 256) {
        const float4 vq = oq4[c4];
        const float4 vk = ok4[c4];
        const float4 py = pY4[c4];
        oqc4[c4] = vq;                 // pass-through copies (unweighted)
        okc4[c4] = vk;

        const float wq[4] = { py.x + 0.1f * (1.0f - py.x),
                              py.y + 0.1f * (1.0f - py.y),
                              py.z + 0.1f * (1.0f - py.z),
                              py.w + 0.1f * (1.0f - py.w) };
        const float sq[4] = { vq.x * wq[0], vq.y * wq[1], vq.z * wq[2], vq.w * wq[3] };
        const float sk[4] = { vk.x * wq[0], vk.y * wq[1], vk.z * wq[2], vk.w * wq[3] };
#pragma unroll
        for (int j = 0; j < 4; ++j) {
            const int c = c4 * 4 + j;
            if (sq[j] > bvq) { bvq = sq[j]; biq = c; }
            if (sk[j] > bvk) { bvk = sk[j]; bik = c; }
        }
    }

    svq[tid] = bvq; siq[tid] = biq;
    svk[tid] = bvk; sik[tid] = bik;
    __syncthreads();

    for (int s = 128; s > 0; s >>= 1) {
        if (tid < s) {
            float v2 = svq[tid + s]; int i2 = siq[tid + s];
            if (v2 > svq[tid] || (v2 == svq[tid] && i2 < siq[tid])) {
                svq[tid] = v2; siq[tid] = i2;
            }
            v2 = svk[tid + s]; i2 = sik[tid + s];
            if (v2 > svk[tid] || (v2 == svk[tid] && i2 < sik[tid])) {
                svk[tid] = v2; sik[tid] = i2;
            }
        }
        __syncthreads();
    }

    if (tid == 0) {
        pseudo_q[b] = siq[0];
        pseudo_k[b] = sik[0];
    }
}

// ---------------------------------------------------------------------------
// Kernel 2: all four logits GEMMs fused. Each wave owns one 16x16 (b,c) tile
// and carries FOUR f32 accumulators (q, k, q_mix, k_mix). The shared
// prototype (B-matrix) fragment is loaded ONCE per K-step and fed to four
// V_WMMA_F32_16X16X4_F32 ops: 5 b64 loads per 4 WMMAs instead of 8 per 4.
//
// f32 WMMA VGPR layouts (ISA 7.12.2):
//   A 16x4 : lanes 0-15 -> M=lane, K={0,1} in v0/v1; lanes 16-31 -> K={2,3}
//   B 4x16 : lanes index N; lane half selects K pair (mirrors A)
//   C/D    : vgpr i: lanes 0-15 -> (M=i, N=lane), lanes 16-31 -> (M=i+8)
// ---------------------------------------------------------------------------
__global__ void __launch_bounds__(128) wmma_logits4_kernel(
    const float* __restrict__ Xq,  const float* __restrict__ Xk,
    const float* __restrict__ Xqm, const float* __restrict__ Xkm,
    const float* __restrict__ P,   // [C_N, D_N] prototypes (L2-resident)
    float* __restrict__ out)       // 4 x [B_N, C_N], consecutive, OC apart
{
    const int lane  = threadIdx.x & 31;
    const int wave  = threadIdx.x >> 5;
    const int ctile = blockIdx.x * 4 + wave;
    if (ctile * 16 >= C_N) return;             // whole-wave exit, EXEC all-1s

    const int c0   = ctile * 16;
    const int b0   = blockIdx.y * 16;
    const int half = lane >> 4;
    const int l16  = lane & 15;

    const size_t aoff = (size_t)(b0 + l16) * D_N + 2 * half;
    const float* aq  = Xq  + aoff;
    const float* ak  = Xk  + aoff;
    const float* aqm = Xqm + aoff;
    const float* akm = Xkm + aoff;
    const float* brow = P + (size_t)(c0 + l16) * D_N + 2 * half;

    v8f acc_q = {}, acc_k = {}, acc_qm = {}, acc_km = {};
#pragma unroll 4
    for (int kk = 0; kk < D_N; kk += 4) {
        const v2f bb = *(const v2f*)(brow + kk);   // shared B fragment
        const v2f a0 = *(const v2f*)(aq  + kk);
        const v2f a1 = *(const v2f*)(ak  + kk);
        const v2f a2 = *(const v2f*)(aqm + kk);
        const v2f a3 = *(const v2f*)(akm + kk);
        acc_q  = __builtin_amdgcn_wmma_f32_16x16x4_f32(false, a0, false, bb, (short)0, acc_q,  false, false);
        acc_k  = __builtin_amdgcn_wmma_f32_16x16x4_f32(false, a1, false, bb, (short)0, acc_k,  false, false);
        acc_qm = __builtin_amdgcn_wmma_f32_16x16x4_f32(false, a2, false, bb, (short)0, acc_qm, false, false);
        acc_km = __builtin_amdgcn_wmma_f32_16x16x4_f32(false, a3, false, bb, (short)0, acc_km, false, false);
    }

    const size_t OC = (size_t)B_N * C_N;
    float* orow = out + (size_t)(b0 + half * 8) * C_N + c0 + l16;
#pragma unroll
    for (int i = 0; i < 8; ++i) {
        orow[(size_t)i * C_N]          = acc_q[i];
        orow[OC     + (size_t)i * C_N] = acc_k[i];
        orow[2 * OC + (size_t)i * C_N] = acc_qm[i];
        orow[3 * OC + (size_t)i * C_N] = acc_km[i];
    }
}

// ---------------------------------------------------------------------------
// Kernel 3: per-class replay of the sequential EMA recurrence + L2 normalize.
// One block per class c, 128 threads = one per feature dim. Labels staged in
// LDS once; the global scan order (all q samples, then all k samples) is
// preserved per class, which is all the recurrence depends on.
// ---------------------------------------------------------------------------
__global__ void __launch_bounds__(128) ema_norm_kernel(
    const float* __restrict__ q,        // [B_N, D_N]
    const float* __restrict__ k,        // [B_N, D_N]
    const float* __restrict__ protos,   // [C_N, D_N]
    const int* __restrict__ pseudo_q,   // [B_N]
    const int* __restrict__ pseudo_k,   // [B_N]
    float* __restrict__ newp)           // [C_N, D_N]
{
    __shared__ int   labs[2 * B_N];
    __shared__ float red[128];

    const int c = blockIdx.x;
    const int d = threadIdx.x;

    for (int i = d; i < B_N; i += 128) {
        labs[i]       = pseudo_q[i];
        labs[B_N + i] = pseudo_k[i];
    }
    __syncthreads();

    float p = protos[(size_t)c * D_N + d];
    for (int i = 0; i < 2 * B_N; ++i) {
        if (labs[i] == c) {                       // uniform (scalar) branch
            const float* f = (i < B_N) ? (q + (size_t)i * D_N)
                                       : (k + (size_t)(i - B_N) * D_N);
            p = 0.99f * p + 0.01f * f[d];
        }
    }

    red[d] = p * p;
    __syncthreads();
    for (int s = 64; s > 0; s >>= 1) {
        if (d < s) red[d] += red[d + s];
        __syncthreads();
    }
    const float nrm = fmaxf(sqrtf(red[0]), 1e-12f);
    newp[(size_t)c * D_N + d] = p / nrm;
}

// ---------------------------------------------------------------------------
// Kernel 4: vectorized pass-through copy of q.
// ---------------------------------------------------------------------------
__global__ void __launch_bounds__(256) copy_q_kernel(
    const float* __restrict__ src, float* __restrict__ dst, int n4)
{
    int i = blockIdx.x * blockDim.x + threadIdx.x;
    if (i < n4) ((float4*)dst)[i] = ((const float4*)src)[i];
}

// ---------------------------------------------------------------------------
extern "C" void kernel_launch(void* const* d_in, const int* in_sizes, int n_in,
                              void* d_out, int out_size, void* d_ws, size_t ws_size,
                              hipStream_t stream) {
    (void)in_sizes; (void)n_in; (void)out_size; (void)ws_size;

    const float* output_q   = (const float*)d_in[0];
    const float* output_k   = (const float*)d_in[1];
    const float* q          = (const float*)d_in[2];
    const float* k          = (const float*)d_in[3];
    const float* q_mix      = (const float*)d_in[4];
    const float* k_mix      = (const float*)d_in[5];
    const float* partial_Y  = (const float*)d_in[6];
    const float* prototypes = (const float*)d_in[7];

    float* out = (float*)d_out;
    const size_t OC = (size_t)B_N * C_N;
    float* out_oq     = out;                                 // output_q copy
    float* out_ok     = out + OC;                            // output_k copy
    float* out_logits = out + 2 * OC;                        // 4 GEMM outputs
    float* out_q      = out + 6 * OC;                        // q copy
    float* out_np     = out + 6 * OC + (size_t)B_N * D_N;    // normalized protos

    int* pseudo_q = (int*)d_ws;
    int* pseudo_k = pseudo_q + B_N;

    // 1) single b128-vectorized pass over the big tensors: copies + argmaxes
    argmax_copy_kernel<<<B_N, 256, 0, stream>>>(
        output_q, output_k, partial_Y, out_oq, out_ok, pseudo_q, pseudo_k);

    // 2) all four WMMA GEMMs fused: shared prototype fragment, 4 accumulators
    dim3 g((C_N / 16 + 3) / 4, B_N / 16);   // (157, 64); 4 waves/block
    wmma_logits4_kernel<<<g, 128, 0, stream>>>(
        q, k, q_mix, k_mix, prototypes, out_logits);

    // 3) q pass-through
    copy_q_kernel<<<((B_N * D_N / 4) + 255) / 256, 256, 0, stream>>>(
        q, out_q, B_N * D_N / 4);

    // 4) EMA recurrence replay per class + normalize (needs pseudo labels)
    ema_norm_kernel<<<C_N, 128, 0, stream>>>(
        q, k, prototypes, pseudo_q, pseudo_k, out_np);
}